// GooLayer_14946486190172
// MI455X (gfx1250) — compile-verified
//
#include <hip/hip_runtime.h>
#include <stdint.h>

// Problem constants (B=2, M=64, D=64, T=8192)
#define BB 2
#define MM 64
#define DD 64
#define TT 8192
#define TILE 32                 // timesteps per LDS tile
#define NTILE (TT / TILE)       // 256 tiles
#define RS 36                   // LDS row stride in dwords (32 data + 4 pad, 16B align)
#define NBUF 4                  // quad-buffered tiles -> tile%4 == buf, unrollable
#define BUFSZ (DD * RS)         // floats per tile buffer
#define HALF (NBUF * BUFSZ)     // h-array offset inside smem (floats)
#define DAMPING 0.9998f

// ---- gfx1250 hardware tanh ------------------------------------------------
__device__ __forceinline__ float htanh(float x) {
#if defined(__has_builtin)
#if __has_builtin(__builtin_amdgcn_tanhf)
    return __builtin_amdgcn_tanhf(x);
#elif __has_builtin(__builtin_amdgcn_tanh_f32)
    return __builtin_amdgcn_tanh_f32(x);
#else
    float y;
    asm volatile("v_tanh_f32 %0, %1\n\ts_nop 1" : "=v"(y) : "v"(x));
    return y;
#endif
#else
    float y;
    asm volatile("v_tanh_f32 %0, %1\n\ts_nop 1" : "=v"(y) : "v"(x));
    return y;
#endif
}

// ---- async copy global -> LDS, saddr form (CDNA5 path, ASYNCcnt) ----------
__device__ __forceinline__ void async_b128_s(uint32_t lds, uint32_t voff,
                                             const float* sbase) {
    asm volatile("global_load_async_to_lds_b128 %0, %1, %2 th:TH_LOAD_NT"
                 :: "v"(lds), "v"(voff), "s"(sbase) : "memory");
}
__device__ __forceinline__ void wait_async_le16() {
    asm volatile("s_wait_asynccnt 0x10" ::: "memory");
}
__device__ __forceinline__ void wait_async_0() {
    asm volatile("s_wait_asynccnt 0x0" ::: "memory");
}

// ---- fused DPP xor-add steps (x += lane_xor(x, N)) -------------------------
template <int CTRL>
__device__ __forceinline__ float xadd_dpp16(float x) {
    int v = __builtin_amdgcn_update_dpp(0, __float_as_int(x), CTRL, 0xF, 0xF, true);
    return x + __int_as_float(v);
}
__device__ __forceinline__ float xadd_dpp8_xor4(float x) {
    constexpr int SEL = (4) | (5 << 3) | (6 << 6) | (7 << 9) |
                        (0 << 12) | (1 << 15) | (2 << 18) | (3 << 21);
    int v = __builtin_amdgcn_mov_dpp8(__float_as_int(x), SEL);
    return x + __int_as_float(v);
}

// Batched xor-reduction of 8 per-lane partials over the 32 lanes of this wave.
// Result: every lane l holds r[t_base + (l & 7)] (replicated per 8-lane group).
__device__ __forceinline__ float reduce8(const float s[8], bool b0, bool b1, bool b2) {
    float a0 = xadd_dpp16<0xB1>(s[0]);
    float a1 = xadd_dpp16<0xB1>(s[1]);
    float a2 = xadd_dpp16<0xB1>(s[2]);
    float a3 = xadd_dpp16<0xB1>(s[3]);
    float a4 = xadd_dpp16<0xB1>(s[4]);
    float a5 = xadd_dpp16<0xB1>(s[5]);
    float a6 = xadd_dpp16<0xB1>(s[6]);
    float a7 = xadd_dpp16<0xB1>(s[7]);
    float m0 = b0 ? a1 : a0;
    float m1 = b0 ? a3 : a2;
    float m2 = b0 ? a5 : a4;
    float m3 = b0 ? a7 : a6;
    m0 = xadd_dpp16<0x4E>(m0);
    m1 = xadd_dpp16<0x4E>(m1);
    m2 = xadd_dpp16<0x4E>(m2);
    m3 = xadd_dpp16<0x4E>(m3);
    float n0 = b1 ? m1 : m0;
    float n1 = b1 ? m3 : m2;
    n0 = xadd_dpp8_xor4(n0);
    n1 = xadd_dpp8_xor4(n1);
    float c = b2 ? n1 : n0;
    c = xadd_dpp16<0x128>(c);          // xor8 (row_ror:8)
    c += __shfl_xor(c, 16, 32);        // xor16
    return c;
}

__global__ __launch_bounds__(64)
void goo_scan_kernel(const float* __restrict__ forces,
                     const float* __restrict__ home,
                     const float* __restrict__ mic,
                     const float* __restrict__ masses,
                     const float* __restrict__ tensions,
                     const float* __restrict__ gains,
                     float* __restrict__ out) {
    __shared__ float smem[2 * HALF];    // [0..HALF): f tiles, [HALF..): h tiles
    __shared__ float rstage[2][TILE];   // wave1 -> wave0 partial handoff

    const int tid  = threadIdx.x;       // 0..63 : 2 waves
    const int lane = tid & 31;
    const int wid  = tid >> 5;
    const int d    = tid;               // this thread's oscillator index (0..63)
    const int chain = blockIdx.x;       // b*64 + m  (128 chains)
    const int m     = chain & (MM - 1);
    const bool b0 = (lane & 1) != 0;
    const bool b1 = (lane & 2) != 0;
    const bool b2 = (lane & 4) != 0;

    const float* fbase = forces + (size_t)chain * DD * TT;
    const float* hbase = home   + (size_t)chain * DD * TT;
    float* outc = out + (size_t)chain * TT;

    const uint32_t ldsBase = (uint32_t)(uintptr_t)(void*)&smem[0];

    const float invm = 1.0f / masses[m];
    const float gD   = gains[m] * DAMPING;          // tanh arg uses u*(g*D)
    const float ks   = tensions[m * DD + d] * invm;
    const float micd = mic[chain * DD + d];

    // Two-step composed recurrence coefficients (per lane, constant).
    // One step: [u,p] <- A[u,p] + [1,D]*c ,  A = [[D,-k],[D^2, 1-D*k]]
    const float Dk   = DAMPING * ks;
    const float d22  = 1.0f - Dk;
    const float tr   = DAMPING + d22;               // a + d
    const float a11  = DAMPING * DAMPING * (1.0f - ks);
    const float a12  = -ks * tr;
    const float a21  = DAMPING * DAMPING * tr;
    const float a22  = d22 * d22 - DAMPING * DAMPING * ks;
    const float q1   = DAMPING * (1.0f - ks);       // A[1,D] (u row)
    const float q2   = DAMPING * tr;                // A[1,D] (p row)
    const float kneg = -ks;

    // Per-thread invariant async addressing: 8 (row, chunk) slots per array.
    uint32_t voff[8];    // global byte offset within chain (sans t0)
    uint32_t ldslo[8];   // LDS byte offset of f-dest (sans buffer offset)
#pragma unroll
    for (int i = 0; i < 8; ++i) {
        const int u  = i * 64 + tid;     // 0..511
        const int dr = u >> 3;           // row 0..63
        const int j  = (u & 7) * 4;      // 0,4,...,28 (floats)
        voff[i]  = (uint32_t)((dr * TT + j) * 4);
        ldslo[i] = ldsBase + (uint32_t)((dr * RS + j) * 4);
    }

    auto issue_tile = [&](int tile, uint32_t bufoff /*bytes*/) {
        const uint32_t tb = (uint32_t)(tile * TILE * 4);
#pragma unroll
        for (int i = 0; i < 8; ++i) {
            const uint32_t vo = voff[i] + tb;
            const uint32_t lf = ldslo[i] + bufoff;
            async_b128_s(lf, vo, fbase);
            async_b128_s(lf + (uint32_t)(HALF * 4), vo, hbase);
        }
    };

    float pos = 0.f, u = 0.f;
    float prev0 = 0.f, prev1 = 0.f, prev2 = 0.f, prev3 = 0.f;  // wave0 partials

    issue_tile(0, 0);
    issue_tile(1, BUFSZ * 4);

// One tile body; J (= tile % 4 = buffer index) is a compile-time constant.
#define TILE_BODY(TIDX, J)                                                      \
    do {                                                                        \
        const int tile_ = (TIDX);                                               \
        if (tile_ == NTILE - 1) wait_async_0();                                 \
        else                    wait_async_le16();                              \
        __syncthreads();                                                        \
        if (tile_ + 2 < NTILE)                                                  \
            issue_tile(tile_ + 2, (uint32_t)((((J) + 2) & 3) * BUFSZ * 4));     \
        if (wid == 0 && tile_ > 0) {                                            \
            const float own = (lane < 16) ? ((lane < 8) ? prev0 : prev1)        \
                                          : ((lane < 24) ? prev2 : prev3);      \
            outc[(tile_ - 1) * TILE + lane] =                                   \
                own + rstage[((J) + 1) & 1][lane];                              \
        }                                                                       \
        const float* fB = &smem[(J) * BUFSZ + d * RS];                          \
        const float* hB = fB + HALF;                                            \
        _Pragma("unroll")                                                       \
        for (int gblk = 0; gblk < 4; ++gblk) {                                  \
            float s[8];                                                         \
            _Pragma("unroll")                                                   \
            for (int q = 0; q < 2; ++q) {                                       \
                const int tt = gblk * 8 + q * 4;                                \
                const float4 F = *(const float4*)&fB[tt];                       \
                const float4 H = *(const float4*)&hB[tt];                       \
                const float* fp = (const float*)&F;                             \
                const float* hp = (const float*)&H;                             \
                _Pragma("unroll")                                               \
                for (int pr = 0; pr < 2; ++pr) {                                \
                    const float c1 = fmaf(hp[2 * pr],     ks, fp[2 * pr]);      \
                    const float c2 = fmaf(hp[2 * pr + 1], ks, fp[2 * pr + 1]);  \
                    const float u1 = fmaf(DAMPING, u, fmaf(kneg, pos, c1));     \
                    const float u2 = fmaf(a11, u,                               \
                                     fmaf(a12, pos, fmaf(q1, c1, c2)));         \
                    const float p2 = fmaf(a21, u, fmaf(a22, pos,                \
                                     fmaf(q2, c1, DAMPING * c2)));              \
                    s[q * 4 + 2 * pr]     = htanh(u1 * gD) * micd;              \
                    s[q * 4 + 2 * pr + 1] = htanh(u2 * gD) * micd;              \
                    u = u2;                                                     \
                    pos = p2;                                                   \
                }                                                               \
            }                                                                   \
            const float r = reduce8(s, b0, b1, b2);                             \
            if (wid == 1) {                                                     \
                if (lane < 8) rstage[(J) & 1][gblk * 8 + lane] = r;             \
            } else {                                                            \
                if      (gblk == 0) prev0 = r;                                  \
                else if (gblk == 1) prev1 = r;                                  \
                else if (gblk == 2) prev2 = r;                                  \
                else                prev3 = r;                                  \
            }                                                                   \
        }                                                                       \
    } while (0)

    for (int big = 0; big < NTILE / 4; ++big) {
        const int t0 = big * 4;
        TILE_BODY(t0 + 0, 0);
        TILE_BODY(t0 + 1, 1);
        TILE_BODY(t0 + 2, 2);
        TILE_BODY(t0 + 3, 3);
    }
#undef TILE_BODY

    // Flush the last tile's results (NTILE-1 has J=3 -> parity 1)
    __syncthreads();
    if (wid == 0) {
        const float own = (lane < 16) ? ((lane < 8) ? prev0 : prev1)
                                      : ((lane < 24) ? prev2 : prev3);
        outc[(NTILE - 1) * TILE + lane] = own + rstage[1][lane];
    }
}

extern "C" void kernel_launch(void* const* d_in, const int* in_sizes, int n_in,
                              void* d_out, int out_size, void* d_ws, size_t ws_size,
                              hipStream_t stream) {
    const float* forces   = (const float*)d_in[0];
    const float* home     = (const float*)d_in[1];
    const float* mic      = (const float*)d_in[2];
    const float* masses   = (const float*)d_in[3];
    const float* tensions = (const float*)d_in[4];
    const float* gains    = (const float*)d_in[5];
    float* out = (float*)d_out;
    (void)in_sizes; (void)n_in; (void)out_size; (void)d_ws; (void)ws_size;

    goo_scan_kernel<<<BB * MM, 64, 0, stream>>>(forces, home, mic, masses,
                                                tensions, gains, out);
}